// Subsampling_71622874628402
// MI455X (gfx1250) — compile-verified
//
#include <hip/hip_runtime.h>

// CDNA5 / gfx1250: wave32, WMMA 16x16x4 F32.
typedef __attribute__((ext_vector_type(2))) float v2f;
typedef __attribute__((ext_vector_type(8))) float v8f;

// Problem constants (from the reference):
//   x:    [32, 64, 256, 256] f32
//   out:  [32, 64, 128, 128] f32
// in-plane  = 256*256 = 65536   (1<<16)
// in-batch  = 64*65536          (1<<22)
// out-plane = 128*128 = 16384   (1<<14)
// out-batch = 64*16384          (1<<20)

__global__ __launch_bounds__(256) void subsample_wmma_kernel(
    const float* __restrict__ x,
    const float* __restrict__ coef,
    const float* __restrict__ bias,
    float* __restrict__ out)
{
    // One wave (32 lanes) produces 32 consecutive output pixels of one output row.
    // 16384 wave-tasks total: 32 batches * 128 rows * 4 groups-of-32.
    const int gwid = blockIdx.x * 8 + (threadIdx.x >> 5);
    const int lane = threadIdx.x & 31;

    const int b   = gwid >> 9;          // 512 wave-tasks per batch
    const int rem = gwid & 511;
    const int y   = rem >> 2;           // output row 0..127
    const int n0  = (rem & 3) << 5;     // output col base: 0,32,64,96

    // B-matrix (4x16 f32, K x N) layout for V_WMMA_F32_16X16X4_F32:
    //   VGPR0: lanes 0-15 = K=0 row, lanes 16-31 = K=2 row
    //   VGPR1: lanes 0-15 = K=1 row, lanes 16-31 = K=3 row
    // We put the 2x2 pooling window in K: K=0,1 = (row 2y, cols 2n,2n+1),
    // K=2,3 = (row 2y+1, cols 2n,2n+1). So each lane needs exactly one b64
    // load of an adjacent float pair: lower half-wave from row 2y, upper
    // half-wave from row 2y+1 -> two contiguous 128B segments per load.
    const int half = lane >> 4;         // 0 -> row 2y (K=0,1); 1 -> row 2y+1 (K=2,3)
    const int lid  = lane & 15;         // N index within the 16-pixel group

    const float* rowp = x + ((size_t)b << 22) + (size_t)(2 * y + half) * 256;
    const int colA = 2 * (n0 + lid);        // outputs n0 .. n0+15
    const int colB = colA + 32;             // outputs n0+16 .. n0+31

    // A = all-ones 16x4 matrix: D[m,n] = sum_k B[k,n] + C[m,n] for every m.
    // Layout-independent since every element is 1.0f.
    const v2f onesA = {1.0f, 1.0f};

    v8f accA = {};   // pooled sums for outputs n0..n0+15   (duplicated rows)
    v8f accB = {};   // pooled sums for outputs n0+16..n0+31

    // 64 channels: one WMMA per channel per 16-output group, accumulation
    // chained through the C operand. Two independent chains interleave, which
    // both hides WMMA->WMMA latency and doubles loads in flight.
    #pragma unroll 8
    for (int c = 0; c < 64; ++c) {
        const float* p = rowp + ((size_t)c << 16);   // next channel plane
        // Read-once streaming input (512 MiB >> 192 MB L2): non-temporal.
        v2f ba = __builtin_nontemporal_load((const v2f*)(p + colA));
        v2f bb = __builtin_nontemporal_load((const v2f*)(p + colB));
        accA = __builtin_amdgcn_wmma_f32_16x16x4_f32(
            false, onesA, false, ba, (short)0, accA, false, false);
        accB = __builtin_amdgcn_wmma_f32_16x16x4_f32(
            false, onesA, false, bb, (short)0, accB, false, false);
    }

    // D row 0: VGPR0 lanes 0-15 = sums[N=0..15]; ones-A duplicates every row,
    // so VGPR0 lanes 16-31 also = sums[N=0..15]. Chain B therefore carries the
    // upper 16 outputs in its upper lane half too -> one select assembles all 32.
    const float pooled = (lane < 16) ? accA.x : accB.x;

    // Broadcast the pooled value to all 64 output channels with per-channel
    // affine. coef/bias are wave-uniform -> scalar loads. Streaming write-once
    // output: non-temporal stores, 128B coalesced per instruction.
    float* obase = out + ((size_t)b << 20) + (size_t)y * 128 + (size_t)(n0 + lane);
    #pragma unroll 8
    for (int c = 0; c < 64; ++c) {
        const float v = fmaf(pooled, coef[c], bias[c]);
        __builtin_nontemporal_store(v, obase + ((size_t)c << 14));
    }
}

extern "C" void kernel_launch(void* const* d_in, const int* in_sizes, int n_in,
                              void* d_out, int out_size, void* d_ws, size_t ws_size,
                              hipStream_t stream)
{
    const float* x    = (const float*)d_in[0];  // [32,64,256,256] f32
    const float* coef = (const float*)d_in[1];  // [64,1,1] f32
    const float* bias = (const float*)d_in[2];  // [64,1,1] f32
    float* out = (float*)d_out;                 // [32,64,128,128] f32

    // 16384 wave-tasks / 8 waves per 256-thread block = 2048 blocks.
    subsample_wmma_kernel<<<dim3(2048), dim3(256), 0, stream>>>(x, coef, bias, out);
}